// LogDet_23424751632853
// MI455X (gfx1250) — compile-verified
//
#include <hip/hip_runtime.h>
#include <hip/hip_bf16.h>
#include <math.h>

// ---------------------------------------------------------------------------
// Batched slogdet of 32768 x (64x64) fp32 matrices + sign-weighted logsumexp.
// One wave (32 lanes) per matrix. Blocked LU (nb=16) in LDS with partial
// pivoting; trailing updates use V_WMMA_F32_16X16X4_F32 (fp32 tensor path).
// Matrix load uses the Tensor Data Mover (TENSOR_LOAD_TO_LDS) with native
// LDS row padding (64 -> 68 floats) to avoid bank conflicts.
// ---------------------------------------------------------------------------

#define NDETS 32768
#define N     64
#define LD    68          // LDS row stride in floats; 68*4=272B (16B aligned rows)

typedef float v2f __attribute__((ext_vector_type(2)));
typedef float v8f __attribute__((ext_vector_type(8)));
typedef unsigned int v4u __attribute__((ext_vector_type(4)));
typedef int v8i __attribute__((ext_vector_type(8)));
typedef int v4i __attribute__((ext_vector_type(4)));

#if defined(__has_builtin)
#if __has_builtin(__builtin_amdgcn_tensor_load_to_lds) && \
    __has_builtin(__builtin_amdgcn_s_wait_tensorcnt)
#define USE_TDM 1
#endif
#endif

__global__ __launch_bounds__(32) void slogdet64_lu_kernel(
    const float* __restrict__ xs,   // [NDETS][64][64] row-major
    float* __restrict__ out_signs,  // [NDETS]
    float* __restrict__ out_logdets // [NDETS]
) {
    __shared__ float As[N * LD];    // ONLY shared object -> LDS offset 0

    const int lane = threadIdx.x;           // 0..31 (wave32)
    const int b    = blockIdx.x;
    const float* G = xs + (size_t)b * (N * N);

#ifdef USE_TDM
    // ---- Tensor Data Mover: one DMA op loads the whole 64x64 tile into
    // LDS, padding each 64-DWORD row with 4 DWORDs (row stride = 68 floats).
    {
        unsigned long long ga = (unsigned long long)(uintptr_t)G;
        v4u g0;
        g0[0] = 1u;                                   // count=1, user D#
        g0[1] = 0u;                                   // lds_addr = 0 (As @ 0)
        g0[2] = (unsigned)(ga & 0xFFFFFFFFu);         // global_addr[31:0]
        g0[3] = ((unsigned)(ga >> 32) & 0x01FFFFFFu)  // global_addr[56:32]
                | (2u << 30);                         // type = 2 ("image")
        v8i g1;
        g1[0] = (2 << 16)      // data_size = 4 bytes
              | (1 << 20)      // pad_enable
              | (5 << 22)      // pad_interval: every 64 DWORDs (one row)
              | (3 << 25);     // pad_amount: 4 DWORDs -> stride 68 floats
        g1[1] = (N << 16);     // tensor_dim0 = 64 (low 16 bits at [63:48])
        g1[2] = (N << 16);     // dim0 hi = 0; tensor_dim1 = 64 (low 16)
        g1[3] = (N << 16);     // dim1 hi = 0; tile_dim0 = 64
        g1[4] = N;             // tile_dim1 = 64; tile_dim2 = 0
        g1[5] = N;             // tensor_dim0_stride = 64 (low 32)
        g1[6] = (4096 << 16);  // stride0 hi = 0; tensor_dim1_stride low16 = 4096
        g1[7] = 0;             // tensor_dim1_stride hi = 0
        v4i g2 = {0, 0, 0, 0};
        v4i g3 = {0, 0, 0, 0};
        v8i g4 = {0, 0, 0, 0, 0, 0, 0, 0};   // 6-arg toolchain: extra group
        __builtin_amdgcn_tensor_load_to_lds(g0, g1, g2, g3, g4, 0);
        __builtin_amdgcn_s_wait_tensorcnt(0);
    }
#else
    // ---- fallback: 1024 float4, perfectly coalesced, via VGPR staging ----
    {
        const float4* G4 = (const float4*)G;
        #pragma unroll
        for (int i = 0; i < 32; ++i) {
            int vid = i * 32 + lane;          // 0..1023
            int row = vid >> 4;               // /16 float4 per row
            int c4  = (vid & 15) << 2;
            float4 v = G4[vid];
            *(float4*)&As[row * LD + c4] = v;
        }
    }
#endif
    __syncthreads();

    int nswap = 0;
    const int half = lane >> 4;     // 0: lanes 0-15, 1: lanes 16-31
    const int l15  = lane & 15;

    for (int kb = 0; kb < N; kb += 16) {
        // ================= panel factorization: cols kb..kb+15 =============
        for (int k = kb; k < kb + 16; ++k) {
            // --- partial pivot search in column k, rows k..63 ---
            float bm = -1.0f; int bi = k;
            for (int i = k + lane; i < N; i += 32) {
                float v = fabsf(As[i * LD + k]);
                if (v > bm) { bm = v; bi = i; }
            }
            #pragma unroll
            for (int off = 16; off > 0; off >>= 1) {
                float om = __shfl_xor(bm, off, 32);
                int   oi = __shfl_xor(bi, off, 32);
                if (om > bm || (om == bm && oi < bi)) { bm = om; bi = oi; }
            }
            const int p = bi;   // uniform across wave
            if (p != k) {
                ++nswap;        // uniform
                // swap full rows k,p: lanes cover cols lane, lane+32
                int c0 = lane, c1 = lane + 32;
                float t0 = As[k * LD + c0];
                As[k * LD + c0] = As[p * LD + c0];
                As[p * LD + c0] = t0;
                float t1 = As[k * LD + c1];
                As[k * LD + c1] = As[p * LD + c1];
                As[p * LD + c1] = t1;
            }
            __syncthreads();
            const float rp = 1.0f / As[k * LD + k];
            // --- scale column k, rank-1 update within panel cols ---
            for (int i = k + 1 + lane; i < N; i += 32) {
                float l = As[i * LD + k] * rp;
                As[i * LD + k] = l;
                float*       Ai = &As[i * LD];
                const float* Ak = &As[k * LD];
                for (int j = k + 1; j < kb + 16; ++j) Ai[j] -= l * Ak[j];
            }
            __syncthreads();
        }
        if (kb == 48) break;   // last block: no trailing part

        // ============== U12 = L11^{-1} * A12 (forward substitution) ========
        for (int m = kb; m < kb + 15; ++m) {
            for (int jj = lane; jj < N - (kb + 16); jj += 32) {
                int   j   = kb + 16 + jj;
                float amj = As[m * LD + j];
                for (int r = m + 1; r < kb + 16; ++r)
                    As[r * LD + j] -= As[r * LD + m] * amj;
            }
            __syncthreads();
        }

        // ============== trailing update: C -= L21 * U12 via WMMA ===========
        // fp32 WMMA 16x16x4; chain 4 per tile for K=16.
        // fp32 WMMA has no A-negate, so load A fragment negated:
        //     D = (-L21) * U12 + C
        const int rem = N - (kb + 16);       // 48, 32, 16
        const int nt  = rem >> 4;            // 3, 2, 1 tiles per dim
        for (int ti = 0; ti < nt; ++ti) {
            const int tm   = kb + 16 + ti * 16;
            const int arow = tm + l15;       // A-matrix: M striped over lanes
            v2f afrag[4];
            #pragma unroll
            for (int q = 0; q < 4; ++q) {
                // lanes 0-15 hold K = 4q+{0,1}; lanes 16-31 hold K = 4q+{2,3}
                int kcol = kb + q * 4 + half * 2;
                afrag[q].x = -As[arow * LD + kcol];
                afrag[q].y = -As[arow * LD + kcol + 1];
            }
            for (int tj = 0; tj < nt; ++tj) {
                const int tn    = kb + 16 + tj * 16;
                const int ccol  = tn + l15;          // N striped over lanes
                const int crow0 = tm + half * 8;     // C/D: VGPR r -> rows r, r+8
                v8f c;
                #pragma unroll
                for (int r = 0; r < 8; ++r) c[r] = As[(crow0 + r) * LD + ccol];
                #pragma unroll
                for (int q = 0; q < 4; ++q) {
                    // B(4x16): VGPR0 -> K rows {0,2}, VGPR1 -> K rows {1,3}
                    int krow = kb + q * 4 + half * 2;
                    v2f bfrag;
                    bfrag.x = As[krow * LD + ccol];
                    bfrag.y = As[(krow + 1) * LD + ccol];
                    c = __builtin_amdgcn_wmma_f32_16x16x4_f32(
                            false, afrag[q], false, bfrag,
                            (short)0, c, false, false);
                }
                #pragma unroll
                for (int r = 0; r < 8; ++r) As[(crow0 + r) * LD + ccol] = c[r];
            }
        }
        __syncthreads();
    }

    // ================= logdet & sign from U diagonal =======================
    float ll = 0.0f, sgn = 1.0f;
    for (int k = lane; k < N; k += 32) {
        float d = As[k * LD + k];
        ll += logf(fabsf(d));
        if (d < 0.0f) sgn = -sgn;
    }
    #pragma unroll
    for (int off = 16; off > 0; off >>= 1) {
        ll  += __shfl_xor(ll,  off, 32);
        sgn *= __shfl_xor(sgn, off, 32);
    }
    if (nswap & 1) sgn = -sgn;
    if (lane == 0) {
        out_signs[b]   = sgn;
        out_logdets[b] = ll;
    }
}

// ---------------------------------------------------------------------------
// Final reduction: logmax, signed sum of exp(logdet - logmax), logpsi, sign.
// Single workgroup; 65536 floats of traffic -> negligible.
// ---------------------------------------------------------------------------
__global__ __launch_bounds__(256) void signed_lse_kernel(
    const float* __restrict__ signs,
    const float* __restrict__ logdets,
    float* __restrict__ out,   // out[0] = logpsi, out[1] = sign_logpsi
    int n)
{
    __shared__ float sm[256];
    const int t = threadIdx.x;

    float m = -INFINITY;
    for (int i = t; i < n; i += 256) m = fmaxf(m, logdets[i]);
    sm[t] = m;
    __syncthreads();
    for (int s = 128; s > 0; s >>= 1) {
        if (t < s) sm[t] = fmaxf(sm[t], sm[t + s]);
        __syncthreads();
    }
    const float logmax = sm[0];
    __syncthreads();

    float acc = 0.0f;
    for (int i = t; i < n; i += 256) acc += signs[i] * expf(logdets[i] - logmax);
    sm[t] = acc;
    __syncthreads();
    for (int s = 128; s > 0; s >>= 1) {
        if (t < s) sm[t] += sm[t + s];
        __syncthreads();
    }
    if (t == 0) {
        float ss = sm[0];
        out[0] = logf(fabsf(ss)) + logmax;
        out[1] = (ss > 0.0f) ? 1.0f : ((ss < 0.0f) ? -1.0f : 0.0f);
    }
}

extern "C" void kernel_launch(void* const* d_in, const int* in_sizes, int n_in,
                              void* d_out, int out_size, void* d_ws, size_t ws_size,
                              hipStream_t stream) {
    (void)in_sizes; (void)n_in; (void)d_ws; (void)ws_size; (void)out_size;

    const float* xs = (const float*)d_in[0];
    float* out      = (float*)d_out;
    // d_out layout: [logpsi, sign_logpsi, signs[NDETS], logdets[NDETS]]
    float* signs    = out + 2;
    float* logdets  = out + 2 + NDETS;

    slogdet64_lu_kernel<<<NDETS, 32, 0, stream>>>(xs, signs, logdets);
    signed_lse_kernel<<<1, 256, 0, stream>>>(signs, logdets, out, NDETS);
}